// QSP_Model_17265768530263
// MI455X (gfx1250) — compile-verified
//
#include <hip/hip_runtime.h>
#include <math.h>

#define LAYERS 64
#define BLOCK  256
#define EPT    4            // elements per thread (x 2 circuits = 8 independent chains)
#define TABLE_F2 130        // 129 (c,s) pairs + 1 pad pair = 1040 bytes = 65 x b128

// ---------------------------------------------------------------------------
// Kernel 1: materialize (cos(phi/2), sin(phi/2)) table in workspace.
// Indices 0..63  : circuit-0 loop layers   (phi[0..63])
//        64      : circuit-0 final         (phi[64])
//        65..127 : circuit-1 loop layers   (phi[65..127])  -- only 63 of them!
//        128     : circuit-1 final         (phi[128])
// ---------------------------------------------------------------------------
__global__ void qsp_prep_coeffs(const float* __restrict__ phi,
                                float* __restrict__ ws) {
    int i = threadIdx.x;
    if (i < 129) {
        float a = 0.5f * phi[i];
        ws[2 * i]     = cosf(a);
        ws[2 * i + 1] = sinf(a);
    } else if (i == 129) {     // pad so the b128 async DMA reads defined data
        ws[258] = 0.0f;
        ws[259] = 0.0f;
    }
}

// One layer: phase rotation by exp(∓i phi/2) on (a2,a3), then the
// data-dependent mix  a2' = x*a2 + i*s*a3 ;  a3' = i*s*a2 + x*a3.
// Scalar-minimal: 8 distinct products + 4 adds per rotation -> 8 mul + 8 fma,
// with all negations folded into FMA source modifiers. 8 independent chains
// per thread give the VOPD dual-issue pairer plenty of parallel FMAs.
__device__ __forceinline__ void qsp_step(float& a2r, float& a2i,
                                         float& a3r, float& a3i,
                                         float pc, float ps,
                                         float x, float s) {
    float b2r = fmaf(a2r, pc,  a2i * ps);
    float b2i = fmaf(a2i, pc, -a2r * ps);
    float b3r = fmaf(a3r, pc, -a3i * ps);
    float b3i = fmaf(a3i, pc,  a3r * ps);
    a2r = fmaf(x, b2r, -s * b3i);
    a2i = fmaf(x, b2i,  s * b3r);
    a3r = fmaf(x, b3r, -s * b2i);
    a3i = fmaf(x, b3i,  s * b2r);
}

// ---------------------------------------------------------------------------
// Kernel 2: main simulation. 8 independent recurrence chains per thread.
// Coefficient table is DMA'd into LDS with the gfx1250 async-to-LDS path,
// overlapped with the per-thread x loads.
// ---------------------------------------------------------------------------
__global__ __launch_bounds__(BLOCK) void qsp_main(
    const float* __restrict__ x_in,
    const float* __restrict__ coef,      // 260 floats in d_ws
    float* __restrict__ out, int n)
{
    __shared__ __align__(16) float2 lut[TABLE_F2];

    // --- CDNA5 async global->LDS copy of the 1040-byte table (65 x b128) ---
    {
        int t = threadIdx.x;
        if (t < 65) {
            const float* g = coef + t * 4;                         // 16B per lane
            auto l3 = (__attribute__((address_space(3))) float2*)(&lut[0]) + 2 * t;
            asm volatile("global_load_async_to_lds_b128 %0, %1, off"
                         :: "v"(l3), "v"(g) : "memory");
        }
    }

    // Issue the x loads while the table DMA is in flight.
    int tid  = blockIdx.x * BLOCK + threadIdx.x;
    int nthr = gridDim.x * BLOCK;

    int   idx[EPT];
    bool  ok[EPT];
    float xr[EPT];
#pragma unroll
    for (int e = 0; e < EPT; ++e) {
        idx[e] = tid + e * nthr;
        ok[e]  = idx[e] < n;
        xr[e]  = ok[e] ? x_in[idx[e]] : 0.0f;
    }

    asm volatile("s_wait_asynccnt 0" ::: "memory");
    __syncthreads();

    float x[EPT], s[EPT];
    float a2r0[EPT], a2i0[EPT], a3r0[EPT], a3i0[EPT];   // circuit 0
    float a2r1[EPT], a2i1[EPT], a3r1[EPT], a3i1[EPT];   // circuit 1

#pragma unroll
    for (int e = 0; e < EPT; ++e) {
        float xv = xr[e];
        x[e] = xv;
        s[e] = sqrtf(fmaxf(fmaf(-xv, xv, 1.0f), 0.0f));   // sin(arccos x)
        a2r0[e] = 0.5f; a2i0[e] = 0.0f; a3r0[e] = 0.5f; a3i0[e] = 0.0f;
        a2r1[e] = 0.5f; a2i1[e] = 0.0f; a3r1[e] = 0.5f; a3i1[e] = 0.0f;
    }

    // 63 layers shared by both circuits (circuit 1 has only 63 loop phis).
#pragma unroll 1
    for (int l = 0; l < LAYERS - 1; ++l) {
        float2 k0 = lut[l];               // circuit 0 layer l   (phi[l])
        float2 k1 = lut[LAYERS + 1 + l];  // circuit 1 layer l   (phi[65+l])
#pragma unroll
        for (int e = 0; e < EPT; ++e) {
            qsp_step(a2r0[e], a2i0[e], a3r0[e], a3i0[e], k0.x, k0.y, x[e], s[e]);
            qsp_step(a2r1[e], a2i1[e], a3r1[e], a3i1[e], k1.x, k1.y, x[e], s[e]);
        }
    }
    {   // layer 63: circuit 0 only
        float2 k0 = lut[LAYERS - 1];
#pragma unroll
        for (int e = 0; e < EPT; ++e)
            qsp_step(a2r0[e], a2i0[e], a3r0[e], a3i0[e], k0.x, k0.y, x[e], s[e]);
    }

    // Final phases; <Z> = Re(a2*e^{-i phif/2} + a3*e^{+i phif/2})
    //             = cf*(a2r+a3r) + sf*(a2i-a3i)
    float2 f0 = lut[LAYERS];        // phi[64]
    float2 f1 = lut[2 * LAYERS];    // phi[128]
#pragma unroll
    for (int e = 0; e < EPT; ++e) {
        float z0 = f0.x * (a2r0[e] + a3r0[e]) + f0.y * (a2i0[e] - a3i0[e]);
        float z1 = f1.x * (a2r1[e] + a3r1[e]) + f1.y * (a2i1[e] - a3i1[e]);
        if (ok[e]) out[idx[e]] = 0.5f * (z0 + z1);
    }
}

extern "C" void kernel_launch(void* const* d_in, const int* in_sizes, int n_in,
                              void* d_out, int out_size, void* d_ws, size_t ws_size,
                              hipStream_t stream) {
    const float* xp  = (const float*)d_in[0];   // data_point [B], f32
    const float* phi = (const float*)d_in[1];   // phi [129], f32
    float* out = (float*)d_out;                 // [B], f32
    float* ws  = (float*)d_ws;                  // >= 1040 bytes used

    int n = in_sizes[0];
    qsp_prep_coeffs<<<1, 256, 0, stream>>>(phi, ws);
    int blocks = (n + BLOCK * EPT - 1) / (BLOCK * EPT);   // 1024 for B = 2^20
    qsp_main<<<blocks, BLOCK, 0, stream>>>(xp, ws, out, n);
}